// LearnableKernel_13443247636677
// MI455X (gfx1250) — compile-verified
//
#include <hip/hip_runtime.h>
#include <hip/hip_bf16.h>
#include <cmath>

// Problem constants (from reference setup_inputs)
#define BATCH 2
#define CCH   128
#define NPTS  3072
#define HEADS 4
#define TOPK  32
#define JCHUNK 512

typedef float v2f __attribute__((ext_vector_type(2)));
typedef float v8f __attribute__((ext_vector_type(8)));

// ---------------------------------------------------------------------------
// Kernel 1: 1x1-conv projections.
//   posT[b][h][n][d] = W_pos[h*16+d] . pos_bot[b,:,n] + b_pos      (d in 0..15)
//   psq[b][h][n]     = || posT[b][h][n][:] ||^2
//   eqF[b][n][d]     = mean_h ( W_val[h*32+d] . corr[b,:,n] + b_val[h*32+d] )
// One thread per (b,n). Weights staged in LDS (value weights pre-averaged
// over heads so we only need 32 combined rows).
// ---------------------------------------------------------------------------
__global__ __launch_bounds__(256) void proj_kernel(
    const float* __restrict__ pos_bot, const float* __restrict__ corr,
    const float* __restrict__ W_pos, const float* __restrict__ b_pos,
    const float* __restrict__ W_val, const float* __restrict__ b_val,
    float* __restrict__ posT, float* __restrict__ psq,
    float* __restrict__ eqF)
{
    __shared__ float sWp[64 * CCH];   // 32 KB
    __shared__ float sWv[32 * CCH];   // 16 KB (head-combined, pre-scaled 1/4)

    const int tid = threadIdx.x;
    for (int idx = tid; idx < 64 * CCH; idx += 256) sWp[idx] = W_pos[idx];
    for (int idx = tid; idx < 32 * CCH; idx += 256) {
        int d = idx / CCH, c = idx % CCH;
        sWv[idx] = 0.25f * (W_val[(0 * 32 + d) * CCH + c] +
                            W_val[(1 * 32 + d) * CCH + c] +
                            W_val[(2 * 32 + d) * CCH + c] +
                            W_val[(3 * 32 + d) * CCH + c]);
    }
    __syncthreads();

    const int gid = blockIdx.x * 256 + tid;           // 0 .. B*N-1
    const int b = gid / NPTS;
    const int n = gid % NPTS;

    const float* pb = pos_bot + (size_t)b * CCH * NPTS + n;
    const float* cf = corr    + (size_t)b * CCH * NPTS + n;

    float accP[64];
    float accV[32];
#pragma unroll
    for (int o = 0; o < 64; ++o) accP[o] = 0.0f;
#pragma unroll
    for (int d = 0; d < 32; ++d) accV[d] = 0.0f;

    for (int c = 0; c < CCH; ++c) {
        const float xp = pb[(size_t)c * NPTS];
        const float xv = cf[(size_t)c * NPTS];
#pragma unroll
        for (int o = 0; o < 64; ++o) accP[o] = fmaf(sWp[o * CCH + c], xp, accP[o]);
#pragma unroll
        for (int d = 0; d < 32; ++d) accV[d] = fmaf(sWv[d * CCH + c], xv, accV[d]);
    }

#pragma unroll
    for (int h = 0; h < HEADS; ++h) {
        float s = 0.0f;
        float* dst = posT + ((size_t)(b * HEADS + h) * NPTS + n) * 16;
#pragma unroll
        for (int d = 0; d < 16; ++d) {
            const float v = accP[h * 16 + d] + b_pos[h * 16 + d];
            dst[d] = v;
            s = fmaf(v, v, s);
        }
        psq[(size_t)(b * HEADS + h) * NPTS + n] = s;
    }

    float* fdst = eqF + ((size_t)b * NPTS + n) * 32;
#pragma unroll
    for (int d = 0; d < 32; ++d) {
        const float bb = 0.25f * (b_val[d] + b_val[32 + d] + b_val[64 + d] + b_val[96 + d]);
        fdst[d] = accV[d] + bb;
    }
}

// ---------------------------------------------------------------------------
// Kernel 2: fused Gram (WMMA f32 16x16x4, K chained 0..15) + per-row
// bottom-32 dist^2 selection + exp + atomic scatter (mean over heads).
// Grid: (N/16, B*H). Block: 512 threads = 16 waves. Wave w computes j-tiles
// {w, w+16} of each 512-column chunk, then scans row w of the chunk.
// ---------------------------------------------------------------------------
__global__ __launch_bounds__(512) void gram_topk_kernel(
    const float* __restrict__ posT, const float* __restrict__ psq,
    const float* __restrict__ beta_param, float* __restrict__ outK)
{
    __shared__ float s_chunk[16 * JCHUNK]; // 32 KB: dist^2 strip, 16 rows x 512 cols
    __shared__ float s_lv[16 * TOPK];      // sorted ascending dist^2 per row
    __shared__ int   s_li[16 * TOPK];      // matching j indices
    __shared__ float s_psqi[16];

    const int tid  = threadIdx.x;
    const int lane = tid & 31;
    const int wv   = tid >> 5;       // wave id 0..15
    const int half = lane >> 4;      // 0 / 1
    const int sub  = lane & 15;      // 0..15

    const int i0 = blockIdx.x * 16;
    const int bh = blockIdx.y;       // b*HEADS + h
    const int b  = bh / HEADS;

    const float* P = posT + (size_t)bh * NPTS * 16;
    const float* Q = psq  + (size_t)bh * NPTS;

    for (int k = tid; k < 16 * TOPK; k += 512) { s_lv[k] = 3.0e38f; s_li[k] = 0; }
    if (tid < 16) s_psqi[tid] = Q[i0 + tid];
    __syncthreads();

    // A fragments for this wave's 16 i-rows, K = 0..15 in 4 steps.
    // ISA layout (A 16x4 f32): lane m = lane%16; VGPR0 = K (half?2:0), VGPR1 = K+1.
    const float* Arow = P + (size_t)(i0 + sub) * 16;
    v2f af[4];
#pragma unroll
    for (int kk = 0; kk < 4; ++kk) {
        af[kk].x = Arow[kk * 4 + 2 * half + 0];
        af[kk].y = Arow[kk * 4 + 2 * half + 1];
    }
    float psqi[8];
#pragma unroll
    for (int r = 0; r < 8; ++r) psqi[r] = s_psqi[r + 8 * half];

    for (int cb = 0; cb < NPTS; cb += JCHUNK) {
        // ---- compute phase: uniform control flow, EXEC all-ones for WMMA ----
#pragma unroll
        for (int t = 0; t < 2; ++t) {
            const int jt = wv + 16 * t;          // tile index in chunk (0..31)
            const int j0 = cb + jt * 16;
            const float* Brow = P + (size_t)(j0 + sub) * 16;
            v8f acc = {};
#pragma unroll
            for (int kk = 0; kk < 4; ++kk) {
                v2f bf;
                bf.x = Brow[kk * 4 + 2 * half + 0];
                bf.y = Brow[kk * 4 + 2 * half + 1];
                acc = __builtin_amdgcn_wmma_f32_16x16x4_f32(
                    /*neg_a=*/false, af[kk], /*neg_b=*/false, bf,
                    /*c_mod=*/(short)0, acc, /*reuse_a=*/false, /*reuse_b=*/false);
            }
            const float psqj = Q[j0 + sub];
            const int jloc = jt * 16 + sub;
#pragma unroll
            for (int r = 0; r < 8; ++r) {
                // C layout: VGPR r -> row (r + 8*half), col = lane%16
                float d2 = psqi[r] + psqj - 2.0f * acc[r];
                d2 = fmaxf(d2, 0.0f);
                s_chunk[(r + 8 * half) * JCHUNK + jloc] = d2;
            }
        }
        __syncthreads();

        // ---- selection phase: wave wv scans row wv (bottom-32 dist^2) ----
        volatile float* lv = s_lv + wv * TOPK;
        volatile int*   li = s_li + wv * TOPK;
        const float* rowv = s_chunk + wv * JCHUNK;
        for (int it = 0; it < JCHUNK / 32; ++it) {
            const float v   = rowv[it * 32 + lane];
            const float thr = lv[TOPK - 1];      // stale read is safe (conservative)
            unsigned mask = __builtin_amdgcn_ballot_w32(v < thr);
            if (lane == 0) {
                while (mask) {
                    const int c = __builtin_ctz(mask);
                    mask &= mask - 1;
                    const float vv = rowv[it * 32 + c];
                    if (vv < lv[TOPK - 1]) {
                        int p = TOPK - 1;
                        while (p > 0 && lv[p - 1] > vv) {
                            lv[p] = lv[p - 1]; li[p] = li[p - 1]; --p;
                        }
                        lv[p] = vv; li[p] = cb + it * 32 + c;
                    }
                }
            }
        }
        __syncthreads();
    }

    // ---- scatter phase: exp only for the <=32 survivors per row/head ----
    const float bp   = beta_param[0];
    const float beta = log1pf(expf(bp));         // softplus
    const float d2   = s_lv[wv * TOPK + lane];
    const int   j    = s_li[wv * TOPK + lane];
    const float val  = 0.25f * expf(-beta * d2); // 1/HEADS fold
    float* dst = outK + (size_t)b * NPTS * NPTS + (size_t)(i0 + wv) * NPTS + j;
    __hip_atomic_fetch_add(dst, val, __ATOMIC_RELAXED, __HIP_MEMORY_SCOPE_AGENT);
}

// ---------------------------------------------------------------------------
extern "C" void kernel_launch(void* const* d_in, const int* in_sizes, int n_in,
                              void* d_out, int out_size, void* d_ws, size_t ws_size,
                              hipStream_t stream) {
    (void)in_sizes; (void)n_in; (void)out_size; (void)ws_size;

    const float* pos_bot    = (const float*)d_in[0];
    const float* corr_feats = (const float*)d_in[1];
    const float* W_pos      = (const float*)d_in[2];
    const float* b_pos      = (const float*)d_in[3];
    const float* W_val      = (const float*)d_in[4];
    const float* b_val      = (const float*)d_in[5];
    const float* beta_param = (const float*)d_in[6];

    float* out  = (float*)d_out;
    float* outK = out;                                          // [B,N,N]
    float* outF = out + (size_t)BATCH * NPTS * NPTS;            // [B,N,32]

    float* posT = (float*)d_ws;                                 // [B,H,N,16]
    float* psq  = posT + (size_t)BATCH * HEADS * NPTS * 16;     // [B,H,N]

    // masked output is sparse (<=128 nonzeros/row): zero-fill then scatter
    hipMemsetAsync(outK, 0, (size_t)BATCH * NPTS * NPTS * sizeof(float), stream);

    proj_kernel<<<(BATCH * NPTS) / 256, 256, 0, stream>>>(
        pos_bot, corr_feats, W_pos, b_pos, W_val, b_val, posT, psq, outF);

    dim3 g2(NPTS / 16, BATCH * HEADS);
    gram_topk_kernel<<<g2, 512, 0, stream>>>(posT, psq, beta_param, outK);
}